// MyGATv2EncoderCategorical_62448824484496
// MI455X (gfx1250) — compile-verified
//
#include <hip/hip_runtime.h>
#include <cmath>

typedef __attribute__((ext_vector_type(2))) float v2f;
typedef __attribute__((ext_vector_type(4))) float v4f;
typedef __attribute__((ext_vector_type(8))) float v8f;

#define LEAKY 0.2f
#define BN_EPS 1e-5f

// ---------------------------------------------------------------------------
// Pad x [N,FIN] -> xp [N,FINP], zero-filled pad columns (GEMM A operand).
// ---------------------------------------------------------------------------
template<int FIN, int FINP>
__global__ void pad_rows(const float* __restrict__ x, float* __restrict__ xp, int N) {
  int tid = blockIdx.x * blockDim.x + threadIdx.x;
  if (tid >= N * FINP) return;
  int n = tid / FINP, c = tid - n * FINP;
  xp[tid] = (c < FIN) ? x[(size_t)n * FIN + c] : 0.f;
}

// att [H,C] -> attp [H,CP], zero pads (keeps score dot fully vectorized).
template<int H, int C>
__global__ void pad_att(const float* __restrict__ att, float* __restrict__ attp) {
  constexpr int CP = (C + 3) & ~3;
  int i = threadIdx.x;
  if (i < H * CP) {
    int h = i / CP, c = i - h * CP;
    attp[i] = (c < C) ? att[h * C + c] : 0.f;
  }
}

// ---------------------------------------------------------------------------
// Dual GEMM (xl = x@Wl, xr = x@Wr) using V_WMMA_F32_16X16X4_F32.
// Output layout is head-padded: [N][H][CP] (CP = ceil(C/4)*4) so downstream
// per-head gathers are 16B-aligned b128 loads. Pad columns get zero weights
// in the LDS stage, so D stores zero-fill them for free.
//  A 16x4 f32 frag: lanes 0-15 K={k0,k0+1}; lanes 16-31 K={k0+2,k0+3}.
//  D 16x16 f32: VGPR v -> M = v + 8*(lane>=16), N = lane&15.
//  LDS B swizzle: element f of (kq,n) = W[kq*4 + (f>>1) + 2*(f&1)][col(n)],
//  so a lane's single ds_load_b64 at offset half*2 yields {W[k0+half], W[k0+half+2]}.
// ---------------------------------------------------------------------------
template<int FIN, int H, int C>
__global__ void __launch_bounds__(256)
gemm_dual(const float* __restrict__ xp,
          const float* __restrict__ Wl, const float* __restrict__ Wr,
          float* __restrict__ xl, float* __restrict__ xr, int N) {
  constexpr int FINP = (FIN + 3) & ~3;
  constexpr int CP   = (C + 3) & ~3;
  constexpr int HP   = H * CP;          // padded output stride
  constexpr int HC   = H * C;           // real weight columns
  constexpr int NT   = (HP + 15) / 16;
  constexpr int NP   = NT * 16;
  constexpr int KQ   = FINP / 4;

  __shared__ float lWl[KQ * NP * 4];
  __shared__ float lWr[KQ * NP * 4];
  for (int i = threadIdx.x; i < KQ * NP * 4; i += blockDim.x) {
    int f  = i & 3;
    int n  = (i >> 2) % NP;
    int kq = i / (4 * NP);
    int k  = kq * 4 + (f >> 1) + 2 * (f & 1);
    int h  = n / CP, c = n - h * CP;
    bool ok = (k < FIN) && (n < HP) && (c < C);
    lWl[i] = ok ? Wl[(size_t)k * HC + h * C + c] : 0.f;
    lWr[i] = ok ? Wr[(size_t)k * HC + h * C + c] : 0.f;
  }
  __syncthreads();

  const int wave = (int)((blockIdx.x * blockDim.x + threadIdx.x) >> 5);
  const int lane = threadIdx.x & 31;
  const int m0 = wave * 16;
  if (m0 < N) {                               // wave-uniform
    const int half = lane >> 4;
    const int l16  = lane & 15;
    const int arow = (m0 + l16 < N) ? (m0 + l16) : (N - 1);

    v8f zero = {};
    v8f accL[NT], accR[NT];
#pragma unroll
    for (int t = 0; t < NT; ++t) { accL[t] = zero; accR[t] = zero; }

    const float* arowp = xp + (size_t)arow * FINP + half * 2;
    for (int kq = 0; kq < KQ; ++kq) {
      v2f a = *(const v2f*)(arowp + kq * 4);
#pragma unroll
      for (int t = 0; t < NT; ++t) {
        const float* base = &lWl[(kq * NP + t * 16 + l16) * 4 + half * 2];
        v2f bl = *(const v2f*)base;
        v2f br = *(const v2f*)(base + (lWr - lWl));
        accL[t] = __builtin_amdgcn_wmma_f32_16x16x4_f32(false, a, false, bl,
                                                        (short)0, accL[t], false, false);
        accR[t] = __builtin_amdgcn_wmma_f32_16x16x4_f32(false, a, false, br,
                                                        (short)0, accR[t], false, false);
      }
    }
#pragma unroll
    for (int t = 0; t < NT; ++t) {
      const int n = t * 16 + l16;
      if (n < HP) {
#pragma unroll
        for (int v = 0; v < 8; ++v) {
          const int row = m0 + v + half * 8;
          if (row < N) {
            xl[(size_t)row * HP + n] = accL[t][v];
            xr[(size_t)row * HP + n] = accR[t][v];
          }
        }
      }
    }
  }
}

// ---------------------------------------------------------------------------
// Segment-softmax attention, stream-ordered phases (head-padded layout)
// ---------------------------------------------------------------------------
__global__ void init_mdenom(float* __restrict__ m, float* __restrict__ den, int NH) {
  int i = blockIdx.x * blockDim.x + threadIdx.x;
  if (i < NH) { m[i] = -INFINITY; den[i] = 0.f; }
}

template<int H, int C>
__global__ void edge_score(const float* __restrict__ xl, const float* __restrict__ xr,
                           const int* __restrict__ src, const int* __restrict__ dst,
                           const float* __restrict__ attp,
                           float* __restrict__ score, float* __restrict__ mbuf, int E) {
  constexpr int CP = (C + 3) & ~3;
  constexpr int HP = H * CP;
  int tid = blockIdx.x * blockDim.x + threadIdx.x;
  if (tid >= E * H) return;
  int e = tid / H, h = tid - e * H;
  int s = src[e], d = dst[e];
  const v4f* pl = (const v4f*)(xl + (size_t)s * HP + h * CP);
  const v4f* pr = (const v4f*)(xr + (size_t)d * HP + h * CP);
  const v4f* pa = (const v4f*)(attp + h * CP);
  float sc = 0.f;
#pragma unroll
  for (int q = 0; q < CP / 4; ++q) {
    v4f v = pl[q] + pr[q];
    v4f a = pa[q];
#pragma unroll
    for (int j = 0; j < 4; ++j) {
      float t = v[j];
      t = (t > 0.f) ? t : LEAKY * t;     // pads contribute leaky(0)*0 = 0
      sc += a[j] * t;
    }
  }
  score[(size_t)e * H + h] = sc;
  __hip_atomic_fetch_max(&mbuf[(size_t)d * H + h], sc,
                         __ATOMIC_RELAXED, __HIP_MEMORY_SCOPE_AGENT);
}

template<int H>
__global__ void edge_exp(const float* __restrict__ mbuf, const int* __restrict__ dst,
                         float* __restrict__ score_p, float* __restrict__ den, int E) {
  int tid = blockIdx.x * blockDim.x + threadIdx.x;
  if (tid >= E * H) return;
  int e = tid / H, h = tid - e * H;
  int d = dst[e];
  float m = mbuf[(size_t)d * H + h];
  if (!__builtin_isfinite(m)) m = 0.f;
  float p = __expf(score_p[(size_t)e * H + h] - m);
  score_p[(size_t)e * H + h] = p;                 // overwrite score with p
  atomicAdd(&den[(size_t)d * H + h], p);
}

template<int H, int C>
__global__ void edge_aggr(const float* __restrict__ p, const float* __restrict__ den,
                          const float* __restrict__ xl,
                          const int* __restrict__ src, const int* __restrict__ dst,
                          float* __restrict__ acc, int E) {
  constexpr int CP = (C + 3) & ~3;
  constexpr int HP = H * CP;
  int tid = blockIdx.x * blockDim.x + threadIdx.x;
  if (tid >= E * H) return;
  int e = tid / H, h = tid - e * H;
  int s = src[e], d = dst[e];
  float alpha = p[(size_t)e * H + h] / (den[(size_t)d * H + h] + 1e-16f);
  const v4f* pl = (const v4f*)(xl + (size_t)s * HP + h * CP);
  float* pa = acc + (size_t)d * HP + h * CP;
  v4f buf[CP / 4];
#pragma unroll
  for (int q = 0; q < CP / 4; ++q) buf[q] = pl[q];   // b128 gathers
#pragma unroll
  for (int c = 0; c < C; ++c)                        // scalar f32 atomics (real chans)
    atomicAdd(&pa[c], alpha * buf[c >> 2][c & 3]);
}

// ---------------------------------------------------------------------------
// BatchNorm over the head-padded accumulator; writes K-padded activation for
// the next layer's GEMM. Channel j maps to padded offset (j/C)*CP + j%C.
// ---------------------------------------------------------------------------
template<int H, int C, int CP>
__global__ void __launch_bounds__(256)
bn_stats(const float* __restrict__ v, const float* __restrict__ bias,
         float* __restrict__ stats, int N) {
  constexpr int HP = H * CP;
  constexpr int D  = H * C;
  constexpr int ROWS = 128;
  __shared__ float ls[2 * D];
  for (int i = threadIdx.x; i < 2 * D; i += blockDim.x) ls[i] = 0.f;
  __syncthreads();
  const int r0 = blockIdx.x * ROWS;
  for (int idx = threadIdx.x; idx < ROWS * D; idx += blockDim.x) {
    int r = r0 + idx / D;
    int j = idx % D;
    if (r < N) {
      int h = j / C, c = j - h * C;
      float val = v[(size_t)r * HP + h * CP + c] + bias[j];
      atomicAdd(&ls[j], val);
      atomicAdd(&ls[D + j], val * val);
    }
  }
  __syncthreads();
  for (int i = threadIdx.x; i < 2 * D; i += blockDim.x) atomicAdd(&stats[i], ls[i]);
}

template<int H, int C, int CP, int FINP, bool ELU>
__global__ void bn_apply(const float* __restrict__ v, const float* __restrict__ bias,
                         const float* __restrict__ stats,
                         const float* __restrict__ gamma, const float* __restrict__ beta,
                         float* __restrict__ out, int N) {
  constexpr int HP = H * CP;
  constexpr int D  = H * C;
  int tid = blockIdx.x * blockDim.x + threadIdx.x;
  if (tid >= N * FINP) return;
  int n = tid / FINP, j = tid - n * FINP;
  float y = 0.f;
  if (j < D) {
    int h = j / C, c = j - h * C;
    float invN = 1.f / (float)N;
    float mu  = stats[j] * invN;
    float var = stats[D + j] * invN - mu * mu;
    float val = v[(size_t)n * HP + h * CP + c] + bias[j];
    y = gamma[j] * (val - mu) * rsqrtf(var + BN_EPS) + beta[j];
    if (ELU) y = (y > 0.f) ? y : (__expf(y) - 1.f);
  }
  out[tid] = y;
}

template<int H, int C>
__global__ void mean_heads(const float* __restrict__ acc, float* __restrict__ v4, int N) {
  constexpr int CP = (C + 3) & ~3;
  constexpr int HP = H * CP;
  int tid = blockIdx.x * blockDim.x + threadIdx.x;
  if (tid >= N * C) return;
  int n = tid / C, c = tid - n * C;
  float s = 0.f;
#pragma unroll
  for (int h = 0; h < H; ++h) s += acc[(size_t)n * HP + h * CP + c];
  v4[tid] = s * (1.f / (float)H);
}

// ---------------------------------------------------------------------------
// Host-side per-layer drivers
// ---------------------------------------------------------------------------
template<int FIN, int H, int C>
static void gat_edges(const float* inPadded, const float* Wl, const float* Wr,
                      const float* att, const int* src, const int* dst, int N, int E,
                      float* xl, float* xr, float* acc, float* sc, float* mbuf, float* den,
                      float* attp, hipStream_t stream) {
  constexpr int CP = (C + 3) & ~3;
  constexpr int HP = H * CP;
  pad_att<H, C><<<1, 64, 0, stream>>>(att, attp);
  int waves = (N + 15) / 16;
  int gGemm = (waves * 32 + 255) / 256;
  gemm_dual<FIN, H, C><<<gGemm, 256, 0, stream>>>(inPadded, Wl, Wr, xl, xr, N);
  int NH = N * H;
  init_mdenom<<<(NH + 255) / 256, 256, 0, stream>>>(mbuf, den, NH);
  hipMemsetAsync(acc, 0, (size_t)N * HP * sizeof(float), stream);
  int EH = E * H;
  int gE = (EH + 255) / 256;
  edge_score<H, C><<<gE, 256, 0, stream>>>(xl, xr, src, dst, attp, sc, mbuf, E);
  edge_exp<H><<<gE, 256, 0, stream>>>(mbuf, dst, sc, den, E);
  edge_aggr<H, C><<<gE, 256, 0, stream>>>(sc, den, xl, src, dst, acc, E);
}

template<int H, int C, int CP, int FINP, bool ELU>
static void bn_run(const float* v, const float* bias, const float* gamma, const float* beta,
                   float* stats, float* out, int N, hipStream_t stream) {
  constexpr int D = H * C;
  hipMemsetAsync(stats, 0, 2 * D * sizeof(float), stream);
  bn_stats<H, C, CP><<<(N + 127) / 128, 256, 0, stream>>>(v, bias, stats, N);
  int NF = N * FINP;
  bn_apply<H, C, CP, FINP, ELU><<<(NF + 255) / 256, 256, 0, stream>>>(v, bias, stats,
                                                                      gamma, beta, out, N);
}

extern "C" void kernel_launch(void* const* d_in, const int* in_sizes, int n_in,
                              void* d_out, int out_size, void* d_ws, size_t ws_size,
                              hipStream_t stream) {
  const float* x  = (const float*)d_in[0];
  const int*   ei = (const int*)d_in[1];
  const int N = in_sizes[0] / 22;
  const int E = in_sizes[1] / 2;
  const int* src = ei;
  const int* dst = ei + E;

  // Workspace layout (floats); all offsets multiples of 4 (16B alignment)
  float* ws = (float*)d_ws;
  size_t o = 0;
  float* xl    = ws + o; o += (size_t)N * 64;   // head-padded [N][H][CP]
  float* xr    = ws + o; o += (size_t)N * 64;
  float* act   = ws + o; o += (size_t)N * 56;   // K-padded activation
  float* acc   = ws + o; o += (size_t)N * 64;   // head-padded accumulator
  float* sc    = ws + o; o += (size_t)E * 5;    // score, then p (in place)
  float* mbuf  = ws + o; o += (size_t)N * 8;
  float* den   = ws + o; o += (size_t)N * 8;
  float* stats = ws + o; o += 128;
  float* v4    = ws + o; o += (size_t)N * 16;   // layer-4 head-mean (dense)
  float* xpad  = ws + o; o += (size_t)N * 24;   // layer-1 K-padded input
  float* attp  = ws + o; o += 64;               // padded attention vector

  // Layer 1: Fin=22(->24), H=3, C=18 (CP=20, HP=60) + BN + ELU  [act stride 56]
  pad_rows<22, 24><<<(N * 24 + 255) / 256, 256, 0, stream>>>(x, xpad, N);
  gat_edges<22, 3, 18>(xpad, (const float*)d_in[2], (const float*)d_in[3],
                       (const float*)d_in[4], src, dst, N, E,
                       xl, xr, acc, sc, mbuf, den, attp, stream);
  bn_run<3, 18, 20, 56, true>(acc, (const float*)d_in[5], (const float*)d_in[6],
                              (const float*)d_in[7], stats, act, N, stream);

  // Layer 2: Fin=54 (stride 56), H=3, C=16 (CP=16, HP=48) + BN  [act stride 48]
  gat_edges<54, 3, 16>(act, (const float*)d_in[8], (const float*)d_in[9],
                       (const float*)d_in[10], src, dst, N, E,
                       xl, xr, acc, sc, mbuf, den, attp, stream);
  bn_run<3, 16, 16, 48, false>(acc, (const float*)d_in[11], (const float*)d_in[12],
                               (const float*)d_in[13], stats, act, N, stream);

  // Layer 3: Fin=48 (stride 48), H=3, C=14 (CP=16, HP=48) + BN  [act stride 44]
  gat_edges<48, 3, 14>(act, (const float*)d_in[14], (const float*)d_in[15],
                       (const float*)d_in[16], src, dst, N, E,
                       xl, xr, acc, sc, mbuf, den, attp, stream);
  bn_run<3, 14, 16, 44, false>(acc, (const float*)d_in[17], (const float*)d_in[18],
                               (const float*)d_in[19], stats, act, N, stream);

  // Layer 4: Fin=42 (stride 44), H=5, C=10 (CP=12, HP=60), mean-heads + BN -> d_out
  gat_edges<42, 5, 10>(act, (const float*)d_in[20], (const float*)d_in[21],
                       (const float*)d_in[22], src, dst, N, E,
                       xl, xr, acc, sc, mbuf, den, attp, stream);
  mean_heads<5, 10><<<(N * 10 + 255) / 256, 256, 0, stream>>>(acc, v4, N);
  bn_run<1, 10, 10, 10, false>(v4, (const float*)d_in[23], (const float*)d_in[24],
                               (const float*)d_in[25], stats, (float*)d_out, N, stream);
}